// InteractionEncoder_18433999635102
// MI455X (gfx1250) — compile-verified
//
#include <hip/hip_runtime.h>
#include <hip/hip_bf16.h>
#include <math.h>

// ---------------------------------------------------------------------------
// InteractionEncoder for MI455X (gfx1250, wave32).
// Kernel 1: one workgroup (256 thr = 8 waves) per bt item.
//   Pass 1 (A=obj/M, B=hum/N): per-human row min+argmin of d^2, all in
//           registers (lane-constant ||h||^2 folded out; ||o||^2 folded into
//           the WMMA K=4 slot as homogeneous coordinate).
//   Pass 2 (A=hum/M, B=obj/N): per-object column min of d^2 as one register
//           per lane; merged with a single shfl_xor(16); plain LDS store.
//   sqrt() only at the per-row feature stage (monotone => same argmin/min).
//   Bitonic sort of dmin_h for quantile features; feats[256][10] -> d_ws.
// Kernel 2: MLP 10->64(relu)->128 via f32 WMMA.
// Dead code in the reference ([:, :10] slice): top-K, mean_rel, mean_dist,
// w_o, dir_o2h, s_o -- all skipped.
// ---------------------------------------------------------------------------

typedef __attribute__((ext_vector_type(2))) float v2f;
typedef __attribute__((ext_vector_type(8))) float v8f;

__device__ __forceinline__ v8f wmma_f32_4(v2f a, v2f b, v8f c) {
#if defined(__HIP_DEVICE_COMPILE__)
  return __builtin_amdgcn_wmma_f32_16x16x4_f32(
      /*neg_a=*/false, a, /*neg_b=*/false, b,
      /*c_mod=*/(short)0, c, /*reuse_a=*/false, /*reuse_b=*/false);
#else
  (void)a; (void)b;
  return c;
#endif
}

#define NPTS 512
#define NTHR 256
#define TAU_INV 20.0f
#define EPS_NORM 1e-6f
#define FBIG 3.402823466e+38f

__device__ __forceinline__ float block_sum(float v, float* s) {
  const int t = threadIdx.x;
  __syncthreads();
  s[t] = v;
  __syncthreads();
  for (int off = NTHR / 2; off > 0; off >>= 1) {
    if (t < off) s[t] += s[t + off];
    __syncthreads();
  }
  return s[0];
}

__device__ __forceinline__ float block_min(float v, float* s) {
  const int t = threadIdx.x;
  __syncthreads();
  s[t] = v;
  __syncthreads();
  for (int off = NTHR / 2; off > 0; off >>= 1) {
    if (t < off) s[t] = fminf(s[t], s[t + off]);
    __syncthreads();
  }
  return s[0];
}

__global__ __launch_bounds__(NTHR)
void ie_features_kernel(const float* __restrict__ human,   // [256,512,3]
                        const float* __restrict__ object,  // [256,512,3]
                        const float* __restrict__ s_h,     // [256,512]
                        float* __restrict__ feats)         // [256,10]
{
  __shared__ float4 sh_h[NPTS];   // x,y,z,|p|^2
  __shared__ float4 sh_o[NPTS];
  __shared__ float  sh_dh[NPTS];  // d^2 min per human, then sorted d
  __shared__ int    sh_ix[NPTS];  // argmin object per human
  __shared__ float  sh_do[NPTS];  // d^2 min per object
  __shared__ float  red[NTHR];

  const int bt   = blockIdx.x;
  const int t    = threadIdx.x;
  const int wv   = t >> 5;
  const int lane = t & 31;
  const int ln   = lane & 15;
  const bool hih = lane >= 16;
  const int hiOff = hih ? 8 : 0;

  const float* hb = human  + (size_t)bt * NPTS * 3;
  const float* ob = object + (size_t)bt * NPTS * 3;

  // ---- stage points + squared norms ----
  for (int r = t; r < NPTS; r += NTHR) {
    float hx = hb[r * 3], hy = hb[r * 3 + 1], hz = hb[r * 3 + 2];
    sh_h[r] = make_float4(hx, hy, hz, hx * hx + hy * hy + hz * hz);
    float ox = ob[r * 3], oy = ob[r * 3 + 1], oz = ob[r * 3 + 2];
    sh_o[r] = make_float4(ox, oy, oz, ox * ox + oy * oy + oz * oz);
  }
  __syncthreads();

  // WMMA f32 16x16x4 operand layout (per lane):
  //   A: lanes 0-15 -> (K0,K1), lanes 16-31 -> (K2,K3); M = lane&15
  //   B: same K split; N = lane&15
  //   C: lanes 0-15: M=v, N=lane; lanes 16-31: M=v+8, N=lane-16
  // Homogeneous trick: A=(x,y,z,|p|^2), B=(-2x',-2y',-2z',1)
  //   => c[v] = |p_M|^2 - 2 p_M . p'_N ;  d^2 = |p'_N|^2 + c[v]

  // ---- pass 1: per-human (N) min+argmin over objects (M) ----
  for (int i = 0; i < 4; ++i) {
    const int n0 = (wv * 4 + i) * 16;
    float4 hp = sh_h[n0 + ln];
    v2f B;
    B.x = hih ? (-2.0f * hp.z) : (-2.0f * hp.x);
    B.y = hih ? 1.0f           : (-2.0f * hp.y);
    float cmin = FBIG;
    int   cidx = 0;
#pragma unroll 4
    for (int mt = 0; mt < 32; ++mt) {
      const int m0 = mt * 16;
      float4 op = sh_o[m0 + ln];
      v2f A;
      A.x = hih ? op.z : op.x;
      A.y = hih ? op.w : op.y;
      v8f c = {};
      c = wmma_f32_4(A, B, c);   // c[v] = oo[m] - 2 o.h,  m = m0+hiOff+v
#pragma unroll
      for (int v = 0; v < 8; ++v) {
        const int m = m0 + hiOff + v;        // ascending in-lane
        if (c[v] < cmin) { cmin = c[v]; cidx = m; }
      }
    }
    // merge half-lanes covering the same human n (complementary m sets)
    float om = __shfl_xor(cmin, 16, 32);
    int   oi = __shfl_xor(cidx, 16, 32);
    if (om < cmin || (om == cmin && oi < cidx)) { cmin = om; cidx = oi; }
    if (!hih) {
      sh_dh[n0 + ln] = hp.w + cmin;   // d^2 (may be tiny-negative)
      sh_ix[n0 + ln] = cidx;
    }
  }

  // ---- pass 2: per-object (N) min over humans (M) ----
  for (int i = 0; i < 4; ++i) {
    const int n0 = (wv * 4 + i) * 16;
    float4 op = sh_o[n0 + ln];
    v2f B;
    B.x = hih ? (-2.0f * op.z) : (-2.0f * op.x);
    B.y = hih ? 1.0f           : (-2.0f * op.y);
    float cmin = FBIG;
#pragma unroll 4
    for (int mt = 0; mt < 32; ++mt) {
      float4 hp = sh_h[mt * 16 + ln];
      v2f A;
      A.x = hih ? hp.z : hp.x;
      A.y = hih ? hp.w : hp.y;
      v8f c = {};
      c = wmma_f32_4(A, B, c);   // c[v] = hh[m] - 2 h.o
#pragma unroll
      for (int v = 0; v < 8; ++v) cmin = fminf(cmin, c[v]);
    }
    cmin = fminf(cmin, __shfl_xor(cmin, 16, 32));
    if (!hih) sh_do[n0 + ln] = op.w + cmin;   // d^2 per object
  }
  __syncthreads();

  // ---- per-row features + block reductions (sqrt only here) ----
  float pd = 0.f, pw = 0.f, px = 0.f, py = 0.f, pz = 0.f, po = 0.f;
  float pmin = FBIG;
  for (int r = t; r < NPTS; r += NTHR) {
    float dm = sqrtf(fmaxf(sh_dh[r], 0.0f));
    int   id = sh_ix[r];
    sh_dh[r] = dm;                 // convert in place for quantile sort
    pd += dm;
    pmin = fminf(pmin, dm);
    pw += expf(-dm * TAU_INV) * s_h[(size_t)bt * NPTS + r];
    float4 hp = sh_h[r];
    float4 op = sh_o[id];
    float vx = op.x - hp.x, vy = op.y - hp.y, vz = op.z - hp.z;
    float inv = 1.0f / sqrtf(fmaxf(vx * vx + vy * vy + vz * vz, EPS_NORM));
    px += vx * inv; py += vy * inv; pz += vz * inv;
    po += sqrtf(fmaxf(sh_do[r], 0.0f));
  }
  const float invN = 1.0f / (float)NPTS;
  float f0 = block_sum(pd, red) * invN;   // mean(dmin_h)
  float f1 = block_min(pmin, red);        // min(dmin_h)
  float f5 = block_sum(pw, red) * invN;   // mean(w_h)
  float f6 = block_sum(px, red) * invN;   // mean dir x
  float f7 = block_sum(py, red) * invN;   // mean dir y
  float f8 = block_sum(pz, red) * invN;   // mean dir z
  float f9 = block_sum(po, red) * invN;   // mean(dmin_o)

  // ---- bitonic sort of dmin_h (512 vals, 256 threads) for quantiles ----
  __syncthreads();
  for (int k = 2; k <= NPTS; k <<= 1) {
    for (int j = k >> 1; j > 0; j >>= 1) {
      int a = ((t / j) * (j << 1)) + (t % j);
      int b = a + j;
      bool up = ((a & k) == 0);
      float va = sh_dh[a], vb = sh_dh[b];
      if (up ? (va > vb) : (va < vb)) { sh_dh[a] = vb; sh_dh[b] = va; }
      __syncthreads();
    }
  }
  // prefix means: kq = round(frac*512) -> 102, 256, 410
  float v1 = sh_dh[t];
  float v2 = sh_dh[t + NTHR];
  float s102 = (t < 102) ? v1 : 0.f;
  float s256 = v1;                          // all t < 256
  float s410 = v1 + ((t + NTHR) < 410 ? v2 : 0.f);
  float q102 = block_sum(s102, red) / 102.0f;
  float q256 = block_sum(s256, red) / 256.0f;
  float q410 = block_sum(s410, red) / 410.0f;

  if (t == 0) {
    float* f = feats + (size_t)bt * 10;
    f[0] = f0; f[1] = f1; f[2] = q102; f[3] = q256; f[4] = q410;
    f[5] = f5; f[6] = f6; f[7] = f7;  f[8] = f8;   f[9] = f9;
  }
}

// ---------------------------------------------------------------------------
// MLP: out = relu(feats @ W1 + b1) @ W2 + b2, all f32 WMMA 16x16x4.
// Grid: 16 blocks x 128 threads (4 waves); block handles 16 rows.
// ---------------------------------------------------------------------------
#define HS 68  // padded hidden row stride (bank-conflict avoidance)

__global__ __launch_bounds__(128)
void ie_mlp_kernel(const float* __restrict__ feats,  // [256,10]
                   const float* __restrict__ W1,     // [10,64]
                   const float* __restrict__ b1,     // [64]
                   const float* __restrict__ W2,     // [64,128]
                   const float* __restrict__ b2,     // [128]
                   float* __restrict__ out)          // [256,128]
{
  __shared__ float hid[16 * HS];

  const int t    = threadIdx.x;
  const int wv   = t >> 5;
  const int lane = t & 31;
  const int ln   = lane & 15;
  const bool hih = lane >= 16;
  const int hiOff = hih ? 8 : 0;
  const int r0   = blockIdx.x * 16;

  // GEMM1: hidden[16x64], wave wv owns column tile n0 = wv*16. K=10 pad to 12.
  {
    const int n = wv * 16 + ln;
    const int row = r0 + ln;
    v8f c = {};
#pragma unroll
    for (int kk = 0; kk < 3; ++kk) {
      const int kb = kk * 4 + (hih ? 2 : 0);
      v2f A, B;
      A.x = (kb     < 10) ? feats[row * 10 + kb]     : 0.0f;
      A.y = (kb + 1 < 10) ? feats[row * 10 + kb + 1] : 0.0f;
      B.x = (kb     < 10) ? W1[kb * 64 + n]          : 0.0f;
      B.y = (kb + 1 < 10) ? W1[(kb + 1) * 64 + n]    : 0.0f;
      c = wmma_f32_4(A, B, c);
    }
    const float bn = b1[n];
#pragma unroll
    for (int v = 0; v < 8; ++v) {
      hid[(v + hiOff) * HS + n] = fmaxf(c[v] + bn, 0.0f);
    }
  }
  __syncthreads();

  // GEMM2: out[16x128], wave wv owns column tiles {wv, wv+4}. K=64.
  for (int ct = wv; ct < 8; ct += 4) {
    const int n = ct * 16 + ln;
    v8f c = {};
#pragma unroll
    for (int kk = 0; kk < 16; ++kk) {
      const int kb = kk * 4 + (hih ? 2 : 0);
      v2f A, B;
      A.x = hid[ln * HS + kb];
      A.y = hid[ln * HS + kb + 1];
      B.x = W2[kb * 128 + n];
      B.y = W2[(kb + 1) * 128 + n];
      c = wmma_f32_4(A, B, c);
    }
    const float bn = b2[n];
#pragma unroll
    for (int v = 0; v < 8; ++v) {
      out[(size_t)(r0 + v + hiOff) * 128 + n] = c[v] + bn;
    }
  }
}

extern "C" void kernel_launch(void* const* d_in, const int* in_sizes, int n_in,
                              void* d_out, int out_size, void* d_ws, size_t ws_size,
                              hipStream_t stream) {
  (void)in_sizes; (void)n_in; (void)out_size; (void)ws_size;
  const float* human  = (const float*)d_in[0];  // [8,32,512,3]
  const float* object = (const float*)d_in[1];  // [8,32,512,3]
  const float* s_h    = (const float*)d_in[2];  // [8,32,512]
  // d_in[3] = s_o : dead code (sliced away by [:, :10])
  const float* W1 = (const float*)d_in[4];
  const float* b1 = (const float*)d_in[5];
  const float* W2 = (const float*)d_in[6];
  const float* b2 = (const float*)d_in[7];
  float* out   = (float*)d_out;
  float* feats = (float*)d_ws;                  // [256,10] scratch

  hipLaunchKernelGGL(ie_features_kernel, dim3(256), dim3(NTHR), 0, stream,
                     human, object, s_h, feats);
  hipLaunchKernelGGL(ie_mlp_kernel, dim3(16), dim3(128), 0, stream,
                     feats, W1, b1, W2, b2, out);
}